// GCPN_33526514712708
// MI455X (gfx1250) — compile-verified
//
#include <hip/hip_runtime.h>
#include <hip/hip_bf16.h>

#define NN   50000
#define NE   800000
#define NTOT 850000   // edges + self loops
#define NG   1000

typedef __attribute__((ext_vector_type(2))) float v2f;
typedef __attribute__((ext_vector_type(8))) float v8f;

__device__ __forceinline__ v8f wmma4(v2f a, v2f b, v8f c) {
  return __builtin_amdgcn_wmma_f32_16x16x4_f32(false, a, false, b, (short)0, c, false, false);
}

__device__ __forceinline__ void atomAddF(float* p, float v) {
  unsafeAtomicAdd(p, v);
}

// ---------------------------------------------------------------------------
// Generic 16-row WMMA tile GEMM: D[16][NT*16] = A[16][K] * W[K][NT*16] + bias
// A-tile in LDS (row stride As), W transposed in LDS: Wt[n*Ws + k] = W[k][n].
// f32 fragment mapping (wave32):
//   A frag: lane(m=lane&15) holds {A[m][k0+2*hi], A[m][k0+1+2*hi]}, hi=lane>>4
//   B frag: lane(n=lane&15) holds {W[k0+2*hi][n], W[k0+1+2*hi][n]}
//   D: vgpr v, lane l -> D[v + 8*(l>>4)][l&15 (+16*nt)]
// ---------------------------------------------------------------------------
template<int NT>
__device__ __forceinline__ void tile_mm_lds(const float* At, int As,
                                            const float* Wt, int Ws,
                                            const float* bias, int ksteps,
                                            int lane, v8f* acc)
{
  const int nl   = lane & 15;
  const int koff = (lane >> 4) << 1;
#pragma unroll
  for (int nt = 0; nt < NT; ++nt) {
    float bv = bias[nt * 16 + nl];
    v8f c;
#pragma unroll
    for (int i = 0; i < 8; ++i) c[i] = bv;
    acc[nt] = c;
  }
  const float* ar = At + nl * As + koff;
  for (int ks = 0; ks < ksteps; ++ks) {
    v2f a = *(const v2f*)(ar + ks * 4);
#pragma unroll
    for (int nt = 0; nt < NT; ++nt) {
      v2f b = *(const v2f*)(Wt + (nt * 16 + nl) * Ws + koff + ks * 4);
      acc[nt] = wmma4(a, b, acc[nt]);
    }
  }
}

// Same but B-fragments read directly from global row-major W[K][Nw]
template<int NT>
__device__ __forceinline__ void tile_mm_gB(const float* At, int As,
                                           const float* __restrict__ Wg, int Nw,
                                           const float* bias, int ksteps,
                                           int lane, v8f* acc)
{
  const int nl   = lane & 15;
  const int koff = (lane >> 4) << 1;
#pragma unroll
  for (int nt = 0; nt < NT; ++nt) {
    float bv = bias[nt * 16 + nl];
    v8f c;
#pragma unroll
    for (int i = 0; i < 8; ++i) c[i] = bv;
    acc[nt] = c;
  }
  const float* ar = At + nl * As + koff;
  for (int ks = 0; ks < ksteps; ++ks) {
    v2f a = *(const v2f*)(ar + ks * 4);
    int kr = ks * 4 + koff;
#pragma unroll
    for (int nt = 0; nt < NT; ++nt) {
      int n = nt * 16 + nl;
      v2f b;
      b.x = Wg[kr * Nw + n];
      b.y = Wg[(kr + 1) * Nw + n];
      acc[nt] = wmma4(a, b, acc[nt]);
    }
  }
}

// ---------------------------------------------------------------------------
__global__ void zero_kernel(float* p, int n) {
  int i = blockIdx.x * blockDim.x + threadIdx.x;
  if (i < n) p[i] = 0.f;
}

// column sums / sumsq for batchnorm: stats[0..63]=sum, [64..127]=sumsq
__global__ __launch_bounds__(256) void bn_stats_kernel(const float* __restrict__ x,
                                                       float* __restrict__ stats, int nrows) {
  __shared__ float ls[128];
  int t = threadIdx.x;
  int col = t & 63, rs = t >> 6;
  if (t < 128) ls[t] = 0.f;
  __syncthreads();
  float s = 0.f, s2 = 0.f;
  for (int r = blockIdx.x * 4 + rs; r < nrows; r += gridDim.x * 4) {
    float v = x[r * 64 + col];
    s += v; s2 += v * v;
  }
  atomicAdd(&ls[col], s);
  atomicAdd(&ls[64 + col], s2);
  __syncthreads();
  if (t < 128) atomAddF(&stats[t], ls[t]);
}

// in-place: x = (x-mean)*gamma*rsqrt(var+eps)+beta  (biased var)
__global__ __launch_bounds__(256) void bn_apply_kernel(float* x, const float* __restrict__ stats,
                                                       const float* __restrict__ gamma,
                                                       const float* __restrict__ beta, int nrows) {
  __shared__ float sc[64], sh[64];
  int t = threadIdx.x;
  if (t < 64) {
    float inv = 1.f / (float)nrows;
    float m = stats[t] * inv;
    float v = stats[64 + t] * inv - m * m;
    float s = gamma[t] * rsqrtf(v + 1e-5f);
    sc[t] = s;
    sh[t] = beta[t] - m * s;
  }
  __syncthreads();
  int total = nrows * 64;
  for (int i = blockIdx.x * blockDim.x + t; i < total; i += gridDim.x * blockDim.x) {
    int c = i & 63;
    x[i] = x[i] * sc[c] + sh[c];
  }
}

// ---------------------------------------------------------------------------
// Edge kernel: per 16-edge tile (1 wave):
//   h = relu([x_dst|x_src|ea] @ A (129x64) + Ab);  w = sigmoid(h @ Bw + Bb)
//   x_prop[dst] += w * x_src   (f32 atomics, L2-resident)
// ---------------------------------------------------------------------------
#define EW 8
__global__ __launch_bounds__(256) void edge_kernel(
    const float* __restrict__ xn, float* __restrict__ xprop,
    const float* __restrict__ AW, const float* __restrict__ Ab,
    const float* __restrict__ BW, const float* __restrict__ Bb,
    const int* __restrict__ src, const int* __restrict__ dst,
    const float* __restrict__ ea, int nE, int nTot)
{
  __shared__ float sAw[64 * 132];               // A^T, zero-padded K rows 129..131
  __shared__ float sAb[64];
  __shared__ float sBw[64];
  __shared__ float sBb;
  __shared__ __align__(16) float sEcat[EW][16 * 132];
  __shared__ int   sSrc[EW][16];
  __shared__ int   sDst[EW][16];
  __shared__ float sGate[EW][16];

  int t = threadIdx.x;
  for (int i = t; i < 129 * 64; i += 256) {
    int k = i >> 6, n = i & 63;
    sAw[n * 132 + k] = AW[i];
  }
  for (int i = t; i < 64 * 3; i += 256) {
    int n = i / 3, k = 129 + (i % 3);
    sAw[n * 132 + k] = 0.f;
  }
  if (t < 64) { sAb[t] = Ab[t]; sBw[t] = BW[t]; }
  if (t == 0) sBb = Bb[0];
  __syncthreads();

  int wave = t >> 5, lane = t & 31;
  int tile = blockIdx.x * EW + wave;
  int e0 = tile * 16;
  bool act = (e0 < nTot);
  int nl = lane & 15, hi = lane >> 4;

  if (lane < 16) {
    int e = e0 + lane;
    int si = 0, di = 0; float ev = 0.f;
    if (act) {
      if (e < nE) { si = src[e]; di = dst[e]; ev = ea[e]; }
      else        { si = di = e - nE; ev = 0.f; }      // self loop, attr 0
    }
    sSrc[wave][lane] = si;
    sDst[wave][lane] = di;
    sEcat[wave][lane * 132 + 128] = ev;
    sEcat[wave][lane * 132 + 129] = 0.f;
    sEcat[wave][lane * 132 + 130] = 0.f;
    sEcat[wave][lane * 132 + 131] = 0.f;
  }
  __builtin_amdgcn_wave_barrier();

  // stage x_dst -> cols 0..63, x_src -> cols 64..127 (float4 gathers, L2 hits)
  for (int i = 0; i < 16; ++i) {
    int flat = i * 32 + lane;          // 0..511
    int half = flat >> 8;              // 0: dst, 1: src
    int r    = (flat >> 4) & 15;
    int c4   = flat & 15;
    int node = half ? sSrc[wave][r] : sDst[wave][r];
    float4 v = make_float4(0.f, 0.f, 0.f, 0.f);
    if (act) v = *(const float4*)(xn + node * 64 + c4 * 4);
    *(float4*)(&sEcat[wave][r * 132 + half * 64 + c4 * 4]) = v;
  }
  __builtin_amdgcn_wave_barrier();

  v8f acc[4];
  tile_mm_lds<4>(&sEcat[wave][0], 132, sAw, 132, sAb, 33, lane, acc);
#pragma unroll
  for (int nt = 0; nt < 4; ++nt)
#pragma unroll
    for (int i = 0; i < 8; ++i) acc[nt][i] = fmaxf(acc[nt][i], 0.f);

  // gate: per-row dot with Bw, butterfly reduce across each 16-lane half
  float p[8];
#pragma unroll
  for (int v = 0; v < 8; ++v) {
    float s = 0.f;
#pragma unroll
    for (int nt = 0; nt < 4; ++nt) s += acc[nt][v] * sBw[nt * 16 + nl];
    p[v] = s;
  }
#pragma unroll
  for (int mask = 1; mask <= 8; mask <<= 1)
#pragma unroll
    for (int v = 0; v < 8; ++v) p[v] += __shfl_xor(p[v], mask, 32);
  if (nl == 0) {
#pragma unroll
    for (int v = 0; v < 8; ++v)
      sGate[wave][hi * 8 + v] = 1.f / (1.f + __expf(-(p[v] + sBb)));
  }
  __builtin_amdgcn_wave_barrier();

  if (act) {
    for (int r = 0; r < 16; ++r) {
      float g = sGate[wave][r];
      int dn = sDst[wave][r];
      float a0 = g * sEcat[wave][r * 132 + 64 + lane];
      float a1 = g * sEcat[wave][r * 132 + 96 + lane];
      atomAddF(&xprop[dn * 64 + lane], a0);
      atomAddF(&xprop[dn * 64 + 32 + lane], a1);
    }
  }
}

// ---------------------------------------------------------------------------
// Node update: out = relu([xn | xprop] (Nx128) @ LW (128x64) + Lb)  (in-place safe)
// ---------------------------------------------------------------------------
__global__ __launch_bounds__(256) void node_kernel(
    const float* xn, const float* __restrict__ xp, float* out,
    const float* __restrict__ LW, const float* __restrict__ Lb, int nNodes)
{
  __shared__ float sWt[64 * 132];
  __shared__ float sLb[64];
  __shared__ __align__(16) float sIn[EW][16 * 132];
  int t = threadIdx.x;
  for (int i = t; i < 128 * 64; i += 256) {
    int k = i >> 6, n = i & 63;
    sWt[n * 132 + k] = LW[i];
  }
  if (t < 64) sLb[t] = Lb[t];
  __syncthreads();

  int wave = t >> 5, lane = t & 31;
  int n0 = (blockIdx.x * EW + wave) * 16;
  bool act = (n0 < nNodes);
  for (int i = 0; i < 16; ++i) {
    int flat = i * 32 + lane;
    int r  = flat >> 5;
    int c4 = flat & 31;
    int node = act ? (n0 + r) : 0;
    const float* sp = (c4 < 16) ? (xn + node * 64 + c4 * 4)
                                : (xp + node * 64 + (c4 - 16) * 4);
    float4 v = act ? *(const float4*)sp : make_float4(0.f, 0.f, 0.f, 0.f);
    *(float4*)(&sIn[wave][r * 132 + c4 * 4]) = v;
  }
  __builtin_amdgcn_wave_barrier();

  v8f acc[4];
  tile_mm_lds<4>(&sIn[wave][0], 132, sWt, 132, sLb, 32, lane, acc);
  if (act) {
    int nl = lane & 15, hi = lane >> 4;
#pragma unroll
    for (int nt = 0; nt < 4; ++nt)
#pragma unroll
      for (int v = 0; v < 8; ++v)
        out[(n0 + v + 8 * hi) * 64 + nt * 16 + nl] = fmaxf(acc[nt][v], 0.f);
  }
}

// Final projection: X = emb @ W(64x64) + b (no relu); in-place safe
__global__ __launch_bounds__(256) void proj_kernel(
    const float* in, float* out,
    const float* __restrict__ W, const float* __restrict__ b, int nNodes)
{
  __shared__ float sWt[64 * 68];
  __shared__ float sB[64];
  __shared__ __align__(16) float sIn[EW][16 * 68];
  int t = threadIdx.x;
  for (int i = t; i < 64 * 64; i += 256) {
    int k = i >> 6, n = i & 63;
    sWt[n * 68 + k] = W[i];
  }
  if (t < 64) sB[t] = b[t];
  __syncthreads();

  int wave = t >> 5, lane = t & 31;
  int n0 = (blockIdx.x * EW + wave) * 16;
  bool act = (n0 < nNodes);
  for (int i = 0; i < 8; ++i) {
    int flat = i * 32 + lane;
    int r = flat >> 4, c4 = flat & 15;
    int node = act ? (n0 + r) : 0;
    float4 v = act ? *(const float4*)(in + node * 64 + c4 * 4) : make_float4(0.f, 0.f, 0.f, 0.f);
    *(float4*)(&sIn[wave][r * 68 + c4 * 4]) = v;
  }
  __builtin_amdgcn_wave_barrier();

  v8f acc[4];
  tile_mm_lds<4>(&sIn[wave][0], 68, sWt, 68, sB, 16, lane, acc);
  if (act) {
    int nl = lane & 15, hi = lane >> 4;
#pragma unroll
    for (int nt = 0; nt < 4; ++nt)
#pragma unroll
      for (int v = 0; v < 8; ++v)
        out[(n0 + v + 8 * hi) * 64 + nt * 16 + nl] = acc[nt][v];
  }
}

// ---------------------------------------------------------------------------
// mf MLP: logit = Wf . relu(W1 . relu(W0 . X + b0) + b1) + bf     (64->128->128->1)
// ---------------------------------------------------------------------------
__global__ __launch_bounds__(256) void mlpf_kernel(
    const float* __restrict__ X, float* __restrict__ logits,
    const float* __restrict__ W0, const float* __restrict__ b0,
    const float* __restrict__ W1, const float* __restrict__ b1,
    const float* __restrict__ Wf, const float* __restrict__ bf, int nNodes)
{
  __shared__ float sW0[128 * 68];
  __shared__ float sb0[128], sb1[128], swf[128];
  __shared__ __align__(16) float sBuf[EW][16 * 132];
  int t = threadIdx.x;
  for (int i = t; i < 64 * 128; i += 256) {
    int k = i >> 7, n = i & 127;
    sW0[n * 68 + k] = W0[i];
  }
  if (t < 128) { sb0[t] = b0[t]; sb1[t] = b1[t]; swf[t] = Wf[t]; }
  __syncthreads();

  int wave = t >> 5, lane = t & 31;
  int nl = lane & 15, hi = lane >> 4;
  int n0 = (blockIdx.x * EW + wave) * 16;
  bool act = (n0 < nNodes);
  for (int i = 0; i < 8; ++i) {
    int flat = i * 32 + lane;
    int r = flat >> 4, c4 = flat & 15;
    int node = act ? (n0 + r) : 0;
    float4 v = act ? *(const float4*)(X + node * 64 + c4 * 4) : make_float4(0.f, 0.f, 0.f, 0.f);
    *(float4*)(&sBuf[wave][r * 132 + c4 * 4]) = v;
  }
  __builtin_amdgcn_wave_barrier();

  v8f h0[8];
  tile_mm_lds<8>(&sBuf[wave][0], 132, sW0, 68, sb0, 16, lane, h0);
  __builtin_amdgcn_wave_barrier();
#pragma unroll
  for (int nt = 0; nt < 8; ++nt)
#pragma unroll
    for (int v = 0; v < 8; ++v)
      sBuf[wave][(v + 8 * hi) * 132 + nt * 16 + nl] = fmaxf(h0[nt][v], 0.f);
  __builtin_amdgcn_wave_barrier();

  v8f h1[8];
  tile_mm_gB<8>(&sBuf[wave][0], 132, W1, 128, sb1, 32, lane, h1);

  float p[8];
#pragma unroll
  for (int v = 0; v < 8; ++v) {
    float s = 0.f;
#pragma unroll
    for (int nt = 0; nt < 8; ++nt) s += fmaxf(h1[nt][v], 0.f) * swf[nt * 16 + nl];
    p[v] = s;
  }
#pragma unroll
  for (int mask = 1; mask <= 8; mask <<= 1)
#pragma unroll
    for (int v = 0; v < 8; ++v) p[v] += __shfl_xor(p[v], mask, 32);
  if (act && nl == 0) {
    float bfv = bf[0];
#pragma unroll
    for (int v = 0; v < 8; ++v) logits[n0 + hi * 8 + v] = p[v] + bfv;
  }
}

// ms MLP on concat[X[a_first[g]], X[n]] (128 -> 128 -> 128 -> 1)
__global__ __launch_bounds__(256) void mlps_kernel(
    const float* __restrict__ X, float* __restrict__ logits,
    const int* __restrict__ actions,
    const float* __restrict__ W0, const float* __restrict__ b0,
    const float* __restrict__ W1, const float* __restrict__ b1,
    const float* __restrict__ Wf, const float* __restrict__ bf, int nNodes)
{
  __shared__ float sW0[128 * 132];
  __shared__ float sb0[128], sb1[128], swf[128];
  __shared__ __align__(16) float sBuf[EW][16 * 132];
  int t = threadIdx.x;
  for (int i = t; i < 128 * 128; i += 256) {
    int k = i >> 7, n = i & 127;
    sW0[n * 132 + k] = W0[i];
  }
  if (t < 128) { sb0[t] = b0[t]; sb1[t] = b1[t]; swf[t] = Wf[t]; }
  __syncthreads();

  int wave = t >> 5, lane = t & 31;
  int nl = lane & 15, hi = lane >> 4;
  int n0 = (blockIdx.x * EW + wave) * 16;
  bool act = (n0 < nNodes);
  for (int i = 0; i < 16; ++i) {
    int flat = i * 32 + lane;
    int r = flat >> 5, c4 = flat & 31;
    int node = act ? (n0 + r) : 0;
    const float* sp;
    if (c4 < 16) {
      int g  = node / 50;
      int af = g * 50 + actions[g * 4];
      sp = X + af * 64 + c4 * 4;
    } else {
      sp = X + node * 64 + (c4 - 16) * 4;
    }
    float4 v = act ? *(const float4*)sp : make_float4(0.f, 0.f, 0.f, 0.f);
    *(float4*)(&sBuf[wave][r * 132 + c4 * 4]) = v;
  }
  __builtin_amdgcn_wave_barrier();

  v8f h0[8];
  tile_mm_lds<8>(&sBuf[wave][0], 132, sW0, 132, sb0, 32, lane, h0);
  __builtin_amdgcn_wave_barrier();
#pragma unroll
  for (int nt = 0; nt < 8; ++nt)
#pragma unroll
    for (int v = 0; v < 8; ++v)
      sBuf[wave][(v + 8 * hi) * 132 + nt * 16 + nl] = fmaxf(h0[nt][v], 0.f);
  __builtin_amdgcn_wave_barrier();

  v8f h1[8];
  tile_mm_gB<8>(&sBuf[wave][0], 132, W1, 128, sb1, 32, lane, h1);

  float p[8];
#pragma unroll
  for (int v = 0; v < 8; ++v) {
    float s = 0.f;
#pragma unroll
    for (int nt = 0; nt < 8; ++nt) s += fmaxf(h1[nt][v], 0.f) * swf[nt * 16 + nl];
    p[v] = s;
  }
#pragma unroll
  for (int mask = 1; mask <= 8; mask <<= 1)
#pragma unroll
    for (int v = 0; v < 8; ++v) p[v] += __shfl_xor(p[v], mask, 32);
  if (act && nl == 0) {
    float bfv = bf[0];
#pragma unroll
    for (int v = 0; v < 8; ++v) logits[n0 + hi * 8 + v] = p[v] + bfv;
  }
}

// ---------------------------------------------------------------------------
// Per-graph heads: softmaxes, me MLP (3-way), mt MLP on mean, X_agg
// ---------------------------------------------------------------------------
__global__ __launch_bounds__(128) void graph_kernel(
    const float* __restrict__ X, const float* __restrict__ flog,
    const float* __restrict__ slog, const int* __restrict__ actions,
    const float* __restrict__ meW0, const float* __restrict__ meb0,
    const float* __restrict__ meW1, const float* __restrict__ meb1,
    const float* __restrict__ meWf, const float* __restrict__ mebf,
    const float* __restrict__ mtW0, const float* __restrict__ mtb0,
    const float* __restrict__ mtW1, const float* __restrict__ mtb1,
    const float* __restrict__ mtWf, const float* __restrict__ mtbf,
    float* __restrict__ out)
{
  __shared__ float sXm[64];
  __shared__ float sIn[128];
  __shared__ float sh0[128];
  __shared__ float sh1[128];
  __shared__ float sef[64], ses[64];
  __shared__ float se[3];
  __shared__ float ssum[2];

  int g = blockIdx.x, t = threadIdx.x;
  int a1 = actions[g * 4 + 0], a2 = actions[g * 4 + 1];
  int ae = actions[g * 4 + 2], ast = actions[g * 4 + 3];
  int base = g * 50;

  if (t < 64) {
    float s = 0.f;
    for (int j = 0; j < 50; ++j) s += X[(base + j) * 64 + t];
    out[4 * NG + g * 64 + t] = s;          // X_agg
    sXm[t] = s * 0.02f;                    // mean
    sef[t] = (t < 50) ? __expf(flog[base + t]) : 0.f;
    ses[t] = (t < 50) ? __expf(slog[base + t]) : 0.f;
  }
  {
    int node = (t < 64) ? (base + a1) : (base + a2);
    sIn[t] = X[node * 64 + (t & 63)];
  }
  __syncthreads();
  if (t == 0) {
    float sf = 0.f, ss = 0.f;
    for (int j = 0; j < 50; ++j) { sf += sef[j]; ss += ses[j]; }
    ssum[0] = sf; ssum[1] = ss;
  }
  // me hidden 0/1
  float a = meb0[t];
  for (int k = 0; k < 128; ++k) a += sIn[k] * meW0[k * 128 + t];
  sh0[t] = fmaxf(a, 0.f);
  __syncthreads();
  a = meb1[t];
  for (int k = 0; k < 128; ++k) a += sh0[k] * meW1[k * 128 + t];
  sh1[t] = fmaxf(a, 0.f);
  __syncthreads();
  if (t < 3) {
    float l = mebf[t];
    for (int k = 0; k < 128; ++k) l += sh1[k] * meWf[k * 3 + t];
    se[t] = __expf(l);
  }
  // mt hidden 0 (from mean)
  a = mtb0[t];
  for (int k = 0; k < 64; ++k) a += sXm[k] * mtW0[k * 128 + t];
  float h0v = fmaxf(a, 0.f);
  __syncthreads();
  sh0[t] = h0v;
  __syncthreads();
  a = mtb1[t];
  for (int k = 0; k < 128; ++k) a += sh0[k] * mtW1[k * 128 + t];
  float h1v = fmaxf(a, 0.f);
  __syncthreads();
  sh1[t] = h1v;
  __syncthreads();
  if (t == 0) {
    float l = mtbf[0];
    for (int k = 0; k < 128; ++k) l += sh1[k] * mtWf[k];
    float pv = 1.f / (1.f + __expf(-l));
    out[g * 4 + 0] = __expf(flog[base + a1]) / ssum[0];
    out[g * 4 + 1] = __expf(slog[base + a2]) / ssum[1];
    out[g * 4 + 2] = se[ae] / (se[0] + se[1] + se[2]);
    out[g * 4 + 3] = ast ? pv : (1.f - pv);
  }
}

// ---------------------------------------------------------------------------
extern "C" void kernel_launch(void* const* d_in, const int* in_sizes, int n_in,
                              void* d_out, int out_size, void* d_ws, size_t ws_size,
                              hipStream_t stream) {
  (void)in_sizes; (void)n_in; (void)out_size; (void)ws_size;
  const float* x       = (const float*)d_in[0];
  const int*   ei      = (const int*)d_in[1];
  const float* ea      = (const float*)d_in[2];
  const int*   actions = (const int*)d_in[4];
  const int* esrc = ei;
  const int* edst = ei + NE;
  auto F = [&](int i) { return (const float*)d_in[i]; };

  float* w     = (float*)d_ws;
  float* xprop = w;                        // NN*64
  float* nb    = w + NN * 64;              // NN*64 (emb, then X in-place)
  float* flog  = w + 2 * NN * 64;          // NN
  float* slog  = flog + NN;                // NN
  float* stats = slog + NN;                // 128

  const int ebx = (NTOT / 16 + EW - 1) / EW;   // 6641
  const int nbx = (NN / 16 + EW - 1) / EW;     // 391
  const int zbx = (NN * 64 + 255) / 256;

  const float* cur = x;
  for (int l = 0; l < 3; ++l) {
    int base = (l == 0) ? 5 : (11 + (l - 1) * 8);
    const float* LW = F(base + 0);
    const float* Lb = F(base + 1);
    const float* AW = F(base + 2);
    const float* Ab = F(base + 3);
    const float* BW = F(base + 4);
    const float* Bb = F(base + 5);
    if (l > 0) {
      const float* gamma = F(base + 6);
      const float* beta  = F(base + 7);
      zero_kernel<<<1, 128, 0, stream>>>(stats, 128);
      bn_stats_kernel<<<256, 256, 0, stream>>>(nb, stats, NN);
      bn_apply_kernel<<<512, 256, 0, stream>>>(nb, stats, gamma, beta, NN);
    }
    zero_kernel<<<zbx, 256, 0, stream>>>(xprop, NN * 64);
    edge_kernel<<<ebx, 256, 0, stream>>>(cur, xprop, AW, Ab, BW, Bb,
                                         esrc, edst, ea, NE, NTOT);
    node_kernel<<<nbx, 256, 0, stream>>>(cur, xprop, nb, LW, Lb, NN);
    cur = nb;
  }
  proj_kernel<<<nbx, 256, 0, stream>>>(nb, nb, F(27), F(28), NN);     // X in-place
  mlpf_kernel<<<nbx, 256, 0, stream>>>(nb, flog, F(29), F(30), F(31), F(32),
                                       F(33), F(34), NN);
  mlps_kernel<<<nbx, 256, 0, stream>>>(nb, slog, actions, F(35), F(36), F(37),
                                       F(38), F(39), F(40), NN);
  graph_kernel<<<NG, 128, 0, stream>>>(nb, flog, slog, actions,
                                       F(41), F(42), F(43), F(44), F(45), F(46),
                                       F(47), F(48), F(49), F(50), F(51), F(52),
                                       (float*)d_out);
}